// Quanv2D_42631845380662
// MI455X (gfx1250) — compile-verified
//
#include <hip/hip_runtime.h>
#include <hip/hip_bf16.h>

// Quanvolution surrogate on gfx1250:
//   out[b, oc, i, j] = 0.1 * precision * sin^2(0.5 * dot(patch(b, oc/4, i, j), weight[oc]))
// Implemented as 4 chained V_WMMA_F32_16X16X4_F32 (K=4 == the 2x2 patch taps,
// f32 A/B keeps the linear stage exact). M = 16 output channels (weights,
// block-diagonal masked per input channel), N = 16 spatial columns (patches).

typedef __attribute__((ext_vector_type(2))) float v2f;
typedef __attribute__((ext_vector_type(8))) float v8f;

#define QV_C   4      // input channels
#define QV_OC  16     // output channels
#define QV_HW  224    // input H == W
#define QV_OHW 223    // output H == W (2x2, stride 1)

__global__ __launch_bounds__(32) void quanv2d_wmma_f32(
    const float* __restrict__ x,      // [B, 4, 224, 224]
    const float* __restrict__ w,      // [16, 4]
    const int*   __restrict__ prec,   // [1]
    float*       __restrict__ out)    // [B, 16, 223, 223]
{
    const int lane = threadIdx.x;      // wave32: 0..31
    const int half = lane >> 4;        // 0: K taps {0,1} / D rows 0..7 ; 1: K taps {2,3} / D rows 8..15
    const int l16  = lane & 15;

    const int bid = blockIdx.x;        // one wave per (b, i) output row
    const int b = bid / QV_OHW;
    const int i = bid % QV_OHW;

    const float scale = 0.1f * (float)prec[0];

    // ---- A operand: weight matrix 16(M=oc) x 4(K), ISA layout:
    //   lane<16 : VGPR0=K0, VGPR1=K1 ; lane>=16 : VGPR0=K2, VGPR1=K3 ; M = lane%16
    // Build 4 channel-masked copies (oc belongs to input channel oc>>2).
    const int oc  = l16;
    const float wa0 = w[oc * 4 + half * 2 + 0];
    const float wa1 = w[oc * 4 + half * 2 + 1];
    const int   cg  = oc >> 2;
    v2f a[QV_C];
#pragma unroll
    for (int c = 0; c < QV_C; ++c) {
        a[c].x = (cg == c) ? wa0 : 0.0f;
        a[c].y = (cg == c) ? wa1 : 0.0f;
    }

    // ---- B operand: patches 4(K) x 16(N=j). ISA layout:
    //   lane<16 : VGPR0=K0, VGPR1=K1 (row i, cols j, j+1)
    //   lane>=16: VGPR0=K2, VGPR1=K3 (row i+1, cols j, j+1)
    const int  row0    = i + half;                                   // i or i+1
    const long xbase_b = (long)b * QV_C * QV_HW * QV_HW;
    const long obase   = (long)b * QV_OC * QV_OHW * QV_OHW + (long)i * QV_OHW;

    for (int j0 = 0; j0 < QV_OHW; j0 += 16) {
        const int jm  = j0 + l16;
        // clamp so the 2-element read stays inside the row span (tail tile only;
        // clamped lanes feed only output columns we never store)
        const int col = (jm > QV_HW - 2) ? (QV_HW - 2) : jm;

        v8f acc = {};
#pragma unroll
        for (int c = 0; c < QV_C; ++c) {
            const long p = xbase_b + ((long)c * QV_HW + row0) * QV_HW + col;
            v2f bm;
            bm.x = x[p];
            bm.y = x[p + 1];
            // prefetch next j-tile of this channel row
            __builtin_prefetch(&x[p + 16], 0, 1);
            // D = A_c x B_c + D  (accumulate all 4 input channels, K total 16)
            acc = __builtin_amdgcn_wmma_f32_16x16x4_f32(
                false, a[c], false, bm, (short)0, acc, false, false);
        }

        // ---- D: VGPR v -> oc = v + 8*half, N = l16 -> column jm.
        // Stores land in contiguous 64B runs along j per (oc, half-wave).
        if (jm < QV_OHW) {
#pragma unroll
            for (int v = 0; v < 8; ++v) {
                const int   occ = v + 8 * half;
                const float s   = __sinf(0.5f * acc[v]);
                out[obase + (long)occ * (QV_OHW * QV_OHW) + jm] = s * s * scale;
            }
        }
    }
}

extern "C" void kernel_launch(void* const* d_in, const int* in_sizes, int n_in,
                              void* d_out, int out_size, void* d_ws, size_t ws_size,
                              hipStream_t stream) {
    const float* x    = (const float*)d_in[0];   // [B,4,224,224] f32
    const float* w    = (const float*)d_in[1];   // [16,4] f32
    const int*   prec = (const int*)d_in[2];     // [1] int
    float*       out  = (float*)d_out;           // [B,16,223,223] f32

    const int B = in_sizes[0] / (QV_C * QV_HW * QV_HW);   // 64 for the reference shapes
    dim3 grid((unsigned)(B * QV_OHW));                    // one wave per (b, i) row
    dim3 block(32);                                       // exactly one wave32
    hipLaunchKernelGGL(quanv2d_wmma_f32, grid, block, 0, stream, x, w, prec, out);
}